// LearnableMedianFilter_64407329571481
// MI455X (gfx1250) — compile-verified
//
#include <hip/hip_runtime.h>
#include <hip/hip_bf16.h>
#include <stdint.h>

// 5x5 median filter, reflect padding, fp32, NCHW 8x3x512x512.
// VALU-bound (~260 v_min/max per pixel); data staged via CDNA5 async
// global->LDS loads (ASYNCcnt) so memory stays off the critical path.

#define TILE    32
#define HALO    2
#define LW      (TILE + 2 * HALO)   // 36: staged tile width/height
#define LSTRIDE 40                  // padded LDS row stride (dwords)
#define IMG     512

__device__ __forceinline__ int reflect512(int p) {
    p = (p < 0) ? -p : p;                      // jnp.pad 'reflect': edge not repeated
    return (p > (IMG - 1)) ? (2 * (IMG - 1) - p) : p;
}

__device__ __forceinline__ void cx(float& a, float& b) {
    float lo = fminf(a, b);
    float hi = fmaxf(a, b);
    a = lo;
    b = hi;
}

// Permutation network: global min -> s[0], global max -> s[N-1],
// all other values preserved (shuffled) in s[1..N-2].
template <int N>
__device__ __forceinline__ void mm(float* s) {
#pragma unroll
    for (int i = 0; i + 1 < N; i += 2) cx(s[i], s[i + 1]);   // pairwise lo/hi
#pragma unroll
    for (int i = 2; i < N; i += 2) cx(s[0], s[i]);           // min funnel (incl. unpaired slot if N odd)
#pragma unroll
    for (int i = 1; i < N - 1; i += 2) cx(s[i], s[N - 1]);   // max funnel
}

__global__ __launch_bounds__(256) void median5x5_kernel(const float* __restrict__ img,
                                                        float* __restrict__ out) {
    __shared__ float tile[LW * LSTRIDE];

    const int tid = threadIdx.x;
    const int x0  = blockIdx.x * TILE;
    const int y0  = blockIdx.y * TILE;
    const float* base  = img + (size_t)blockIdx.z * (IMG * IMG);
    float*       obase = out + (size_t)blockIdx.z * (IMG * IMG);

    // ---- Stage 36x36 reflected tile into LDS via async global->LDS loads ----
    const unsigned lds_base = (unsigned)(uintptr_t)tile;  // low 32 bits = LDS byte offset
    for (int i = tid; i < LW * LW; i += 256) {
        int ly = i / LW;
        int lx = i - ly * LW;
        int gx = reflect512(x0 - HALO + lx);
        int gy = reflect512(y0 - HALO + ly);
        const float* src = base + gy * IMG + gx;
        unsigned dst = lds_base + (unsigned)(ly * LSTRIDE + lx) * 4u;
        asm volatile("global_load_async_to_lds_b32 %0, %1, off"
                     :
                     : "v"(dst), "v"(src)
                     : "memory");
    }
    asm volatile("s_wait_asynccnt 0x0" ::: "memory");  // this wave's async loads landed in LDS
    __syncthreads();                                   // all waves' tiles visible

    // ---- Each thread: 4 output pixels (rows oyb, oyb+8, oyb+16, oyb+24) ----
    const int ox  = tid & (TILE - 1);
    const int oyb = tid >> 5;

#pragma unroll 1
    for (int k = 0; k < 4; ++k) {
        const int oy = oyb + k * 8;
        const float* w = &tile[oy * LSTRIDE + ox];
#define W(dy, dx) w[(dy) * LSTRIDE + (dx)]

        // Forgetful selection for median of 25 (13th smallest):
        // working set of 14 -> eject min+max -> insert next -> repeat.
        float a[14];
        a[0]  = W(0, 0); a[1]  = W(0, 1); a[2]  = W(0, 2); a[3]  = W(0, 3); a[4]  = W(0, 4);
        a[5]  = W(1, 0); a[6]  = W(1, 1); a[7]  = W(1, 2); a[8]  = W(1, 3); a[9]  = W(1, 4);
        a[10] = W(2, 0); a[11] = W(2, 1); a[12] = W(2, 2); a[13] = W(2, 3);
        mm<14>(a);
        a[0] = W(2, 4); mm<13>(a);
        a[0] = W(3, 0); mm<12>(a);
        a[0] = W(3, 1); mm<11>(a);
        a[0] = W(3, 2); mm<10>(a);
        a[0] = W(3, 3); mm<9>(a);
        a[0] = W(3, 4); mm<8>(a);
        a[0] = W(4, 0); mm<7>(a);
        a[0] = W(4, 1); mm<6>(a);
        a[0] = W(4, 2); mm<5>(a);
        a[0] = W(4, 3); mm<4>(a);
        a[0] = W(4, 4);
        // 3 survivors: median = middle
        float mn  = fminf(a[0], a[1]);
        float mx  = fmaxf(a[0], a[1]);
        float med = fmaxf(mn, fminf(mx, a[2]));

        obase[(size_t)(y0 + oy) * IMG + (x0 + ox)] = med;
#undef W
    }
}

extern "C" void kernel_launch(void* const* d_in, const int* in_sizes, int n_in,
                              void* d_out, int out_size, void* d_ws, size_t ws_size,
                              hipStream_t stream) {
    const float* img = (const float*)d_in[0];
    float* out = (float*)d_out;

    const int planes = in_sizes[0] / (IMG * IMG);  // B*C = 24
    dim3 grid(IMG / TILE, IMG / TILE, planes);
    median5x5_kernel<<<grid, 256, 0, stream>>>(img, out);
}